// LatentClassifier_45174466019695
// MI455X (gfx1250) — compile-verified
//
#include <hip/hip_runtime.h>
#include <math.h>
#include <stdint.h>

// ---------------------------------------------------------------------------
// LatentClassifier collapses (size-1 layernorms emit their biases exactly;
// the S x S sgu einsums multiply exact zeros) to:
//   c[s]   = sum_i cp2_w[i]*gelu(cp1_b[i,0])*sgu_proj_b[i,s] + cp2_b[i]
//   q[n]   = sum_s c[s]*pooler_w[n,s]            (precomputed)
//   z[b,n] = gelu( x @ pooler_w^T + pooler_b + q )
//   out[b] = sum_n z[b,n]*cls_w[n] + cls_b
// Only real work: one 4096^3 fp32 GEMM -> f32 WMMA 16x16x4.
// Staging: double-buffered async global->LDS (ASYNCcnt), both tiles row-major
// with 36-float row stride (16B-aligned b128 copies, conflict-free b64 frags).
// ---------------------------------------------------------------------------

typedef __attribute__((ext_vector_type(2))) float v2f;
typedef __attribute__((ext_vector_type(8))) float v8f;

#define SDIM 4096
#define BM 256
#define BN 128
#define BK 32
#define NKB (SDIM / BK)
#define ASTR 36            // row stride (floats): b128-alignable, bank-clean

__device__ __forceinline__ float gelu_exact(float v) {
    return 0.5f * v * (1.0f + erff(v * 0.70710678118654752440f));
}

// c[s] into ws, out[b] = cls_b[0]
__global__ void lc_prep1(const float* __restrict__ cp1_b,
                         const float* __restrict__ cp2_w,
                         const float* __restrict__ cp2_b,
                         const float* __restrict__ sgu_proj_b,
                         const float* __restrict__ cls_b,
                         float* __restrict__ c,
                         float* __restrict__ out) {
    int s = blockIdx.x * blockDim.x + threadIdx.x;
    if (s < SDIM) {
        float u0 = gelu_exact(cp1_b[0]);   // layer 0, ch 0 (D_FFN stride 2)
        float u1 = gelu_exact(cp1_b[2]);   // layer 1, ch 0
        c[s] = u0 * cp2_w[0] * sgu_proj_b[s]        + cp2_b[0]
             + u1 * cp2_w[1] * sgu_proj_b[SDIM + s] + cp2_b[1];
        out[s] = cls_b[0];                  // B == S == 4096
    }
}

// pbq[n] = pooler_b[n] + sum_s c[s] * pooler_w[n,s]   (one wave per n)
__global__ void lc_prep2(const float* __restrict__ w,
                         const float* __restrict__ pb,
                         const float* __restrict__ c,
                         float* __restrict__ pbq) {
    const int wave = threadIdx.x >> 5;
    const int lane = threadIdx.x & 31;
    const int n = blockIdx.x * 8 + wave;
    const float* row = w + (size_t)n * SDIM;
    float sum = 0.0f;
    for (int s = lane * 4; s < SDIM; s += 128) {
        const float4 wv = *reinterpret_cast<const float4*>(row + s);
        const float4 cv = *reinterpret_cast<const float4*>(c + s);
        sum += wv.x * cv.x + wv.y * cv.y + wv.z * cv.z + wv.w * cv.w;
    }
    sum += __shfl_xor(sum, 1, 32);
    sum += __shfl_xor(sum, 2, 32);
    sum += __shfl_xor(sum, 4, 32);
    sum += __shfl_xor(sum, 8, 32);
    sum += __shfl_xor(sum, 16, 32);
    if (lane == 0) pbq[n] = pb[n] + sum;
}

// Shared IOFFSET is added to BOTH the memory and LDS addresses; rows are
// copied whole, so within-row offsets advance both sides identically.
#define ASYNC_CP(LDSREG, VOFF, SREG, OFF)                                   \
    asm volatile("global_load_async_to_lds_b128 %0, %1, %2 offset:" #OFF    \
                 :: "v"(LDSREG), "v"(VOFF), "s"(SREG) : "memory")

// 8 b128 rows-chunks for A (256 rows, 1 thread/row) + 4 for B (128 rows,
// 2 threads/row), then slide the global byte offsets to the next K-block.
#define ISSUE(BUF)                                                          \
    do {                                                                    \
        ASYNC_CP(a_ldsb[BUF], xoff, xp, 0);                                 \
        ASYNC_CP(a_ldsb[BUF], xoff, xp, 16);                                \
        ASYNC_CP(a_ldsb[BUF], xoff, xp, 32);                                \
        ASYNC_CP(a_ldsb[BUF], xoff, xp, 48);                                \
        ASYNC_CP(a_ldsb[BUF], xoff, xp, 64);                                \
        ASYNC_CP(a_ldsb[BUF], xoff, xp, 80);                                \
        ASYNC_CP(a_ldsb[BUF], xoff, xp, 96);                                \
        ASYNC_CP(a_ldsb[BUF], xoff, xp, 112);                               \
        ASYNC_CP(b_ldsb[BUF], woff, wp, 0);                                 \
        ASYNC_CP(b_ldsb[BUF], woff, wp, 16);                                \
        ASYNC_CP(b_ldsb[BUF], woff, wp, 32);                                \
        ASYNC_CP(b_ldsb[BUF], woff, wp, 48);                                \
        xoff += BK * 4;                                                     \
        woff += BK * 4;                                                     \
    } while (0)

// 8 K-steps of 4: 4 A-frag + 4 B-frag b64 LDS reads, 16 f32 WMMAs each.
#define COMPUTE_TILE(BUF)                                                   \
    do {                                                                    \
        _Pragma("unroll")                                                   \
        for (int ks = 0; ks < BK; ks += 4) {                                \
            v2f a[4], b[4];                                                 \
            _Pragma("unroll")                                               \
            for (int sm = 0; sm < 4; ++sm)                                  \
                a[sm] = *reinterpret_cast<const v2f*>(                      \
                    &As[BUF][wm * 64 + sm * 16 + l16][ks + 2 * kh]);        \
            _Pragma("unroll")                                               \
            for (int sn = 0; sn < 4; ++sn)                                  \
                b[sn] = *reinterpret_cast<const v2f*>(                      \
                    &Bs[BUF][wn * 64 + sn * 16 + l16][ks + 2 * kh]);        \
            _Pragma("unroll")                                               \
            for (int sm = 0; sm < 4; ++sm)                                  \
                _Pragma("unroll")                                           \
                for (int sn = 0; sn < 4; ++sn)                              \
                    acc[sm][sn] = __builtin_amdgcn_wmma_f32_16x16x4_f32(    \
                        false, a[sm], false, b[sn],                         \
                        (short)0, acc[sm][sn], false, false);               \
        }                                                                   \
    } while (0)

__global__ __launch_bounds__(256, 1)
void lc_gemm_kernel(const float* __restrict__ xp,   // (B,S)
                    const float* __restrict__ wp,   // pooler_w (S,S)
                    const float* __restrict__ pbq,  // (S,) pooler_b + q
                    const float* __restrict__ cw,   // cls_w (1,S)
                    float* __restrict__ out) {      // (B,)
    __shared__ float As[2][BM][ASTR];   // 2 x 36 KB
    __shared__ float Bs[2][BN][ASTR];   // 2 x 18 KB  (row-major, same as memory)

    const int tid  = threadIdx.x;
    const int lane = tid & 31;
    const int wave = tid >> 5;       // 0..7
    const int wm   = wave >> 1;      // 0..3 -> rows wm*64
    const int wn   = wave & 1;       // 0..1 -> cols wn*64
    const int l16  = lane & 15;
    const int kh   = lane >> 4;      // K half per ISA A/B fragment layout
    const int bm0  = blockIdx.y * BM;
    const int bn0  = blockIdx.x * BN;

    v8f acc[4][4];
    const v8f vzero = {0.f, 0.f, 0.f, 0.f, 0.f, 0.f, 0.f, 0.f};
#pragma unroll
    for (int i = 0; i < 4; ++i)
#pragma unroll
        for (int j = 0; j < 4; ++j) acc[i][j] = vzero;

    // ---- async staging addresses --------------------------------------
    const int nB     = tid >> 1;     // 0..127 : pooler_w row
    const int khalfB = tid & 1;      // which 16-float half of the B row

    const unsigned a_base = (unsigned)(uintptr_t)&As[0][0][0];
    const unsigned b_base = (unsigned)(uintptr_t)&Bs[0][0][0];
    const unsigned a_ldsb[2] = {
        a_base + (unsigned)tid * (ASTR * 4u),
        a_base + (unsigned)(BM * ASTR * 4) + (unsigned)tid * (ASTR * 4u) };
    const unsigned b_ldsb[2] = {
        b_base + (unsigned)(nB * ASTR + khalfB * 16) * 4u,
        b_base + (unsigned)(BN * ASTR * 4) + (unsigned)(nB * ASTR + khalfB * 16) * 4u };

    unsigned xoff = (unsigned)(bm0 + tid) * (SDIM * 4u);
    unsigned woff = (unsigned)(bn0 + nB) * (SDIM * 4u) + (unsigned)(khalfB * 64);

    // ---- software pipeline: copy K-block kb+1 while computing kb ------
    ISSUE(0);                                   // K-block 0 -> buffer 0
#pragma unroll 1
    for (int kb = 0; kb < NKB; kb += 2) {
        asm volatile("s_wait_asynccnt 0x0" ::: "memory");
        __syncthreads();                        // buf0 ready; buf1 reads retired
        ISSUE(1);                               // K-block kb+1 -> buffer 1
        COMPUTE_TILE(0);

        asm volatile("s_wait_asynccnt 0x0" ::: "memory");
        __syncthreads();                        // buf1 ready; buf0 reads retired
        if (kb + 2 < NKB) ISSUE(0);             // K-block kb+2 -> buffer 0
        COMPUTE_TILE(1);
    }

    // ---- fused epilogue: +pbq, gelu, *cls_w, reduce over n, row atomics ----
    float pbv[4], cwv[4];
#pragma unroll
    for (int sn = 0; sn < 4; ++sn) {
        const int gn = bn0 + wn * 64 + sn * 16 + l16;
        pbv[sn] = pbq[gn];
        cwv[sn] = cw[gn];
    }
#pragma unroll
    for (int sm = 0; sm < 4; ++sm) {
#pragma unroll
        for (int r = 0; r < 8; ++r) {
            float sum = 0.0f;
#pragma unroll
            for (int sn = 0; sn < 4; ++sn)
                sum += gelu_exact(acc[sm][sn][r] + pbv[sn]) * cwv[sn];
            // C/D layout: lanes 0-15 -> M=r, lanes 16-31 -> M=r+8
            sum += __shfl_xor(sum, 1, 32);
            sum += __shfl_xor(sum, 2, 32);
            sum += __shfl_xor(sum, 4, 32);
            sum += __shfl_xor(sum, 8, 32);
            if (l16 == 0) {
                const int m = bm0 + wm * 64 + sm * 16 + r + 8 * kh;
                atomicAdd(out + m, sum);
            }
        }
    }
}

extern "C" void kernel_launch(void* const* d_in, const int* in_sizes, int n_in,
                              void* d_out, int out_size, void* d_ws, size_t ws_size,
                              hipStream_t stream) {
    (void)in_sizes; (void)n_in; (void)out_size; (void)ws_size;
    const float* x          = (const float*)d_in[0];
    const float* cp1_b      = (const float*)d_in[4];
    const float* sgu_proj_b = (const float*)d_in[8];
    const float* cp2_w      = (const float*)d_in[9];
    const float* cp2_b      = (const float*)d_in[10];
    const float* pooler_w   = (const float*)d_in[11];
    const float* pooler_b   = (const float*)d_in[12];
    const float* cls_w      = (const float*)d_in[13];
    const float* cls_b      = (const float*)d_in[14];

    float* out = (float*)d_out;
    float* c   = (float*)d_ws;          // 4096 floats
    float* pbq = (float*)d_ws + SDIM;   // 4096 floats

    lc_prep1<<<dim3(SDIM / 256), dim3(256), 0, stream>>>(
        cp1_b, cp2_w, cp2_b, sgu_proj_b, cls_b, c, out);

    lc_prep2<<<dim3(SDIM / 8), dim3(256), 0, stream>>>(
        pooler_w, pooler_b, c, pbq);

    lc_gemm_kernel<<<dim3(SDIM / BN, SDIM / BM), dim3(256), 0, stream>>>(
        x, pooler_w, pbq, cls_w, out);
}